// MoEAllReduce_57604101374328
// MI455X (gfx1250) — compile-verified
//
#include <hip/hip_runtime.h>
#include <hip/hip_bf16.h>
#include <stdint.h>

// ---------------------------------------------------------------------------
// MoE expert-reduction + residual add + RMSNorm, MI455X (gfx1250).
// Memory-bound (~1.65 GB @ 23.3 TB/s ~= 70us floor). One block per token row.
// gfx1250 paths: global_load_async_to_lds_b128 (ASYNCcnt) double-buffer
// pipeline for the 1 GiB expert stream, b128 non-temporal loads/stores for
// all streamed tensors, wave32 shuffle reduction, split s_wait_* counters.
// WMMA is inapplicable: the e-contraction has a per-token scale vector, so
// it is a diagonal-scaled sum (0.33 FLOP/byte), not a GEMM.
// ---------------------------------------------------------------------------

typedef float __attribute__((ext_vector_type(4))) fvec4;
typedef int   __attribute__((ext_vector_type(4))) iv4;

#define AS1 __attribute__((address_space(1)))
#define AS3 __attribute__((address_space(3)))

constexpr int   kH      = 4096;          // hidden size (reference fixes this)
constexpr int   kBlock  = 256;           // 8 wave32 per block
constexpr int   kNV     = kH / 4;        // 1024 float4 per row
constexpr int   kChunks = kNV / kBlock;  // 4 float4 per thread per tensor
constexpr int   kWaves  = kBlock / 32;
constexpr float kEps    = 1e-5f;

#if defined(__gfx1250__) && __has_builtin(__builtin_amdgcn_global_load_async_to_lds_b128)
#define USE_ASYNC_LDS 1
#else
#define USE_ASYNC_LDS 0
#endif

#if defined(__gfx1250__)
#if __has_builtin(__builtin_amdgcn_s_wait_asynccnt)
#define WAIT_ASYNC(n) __builtin_amdgcn_s_wait_asynccnt(n)
#else
#define WAIT_ASYNC(n) asm volatile("s_wait_asynccnt %0" ::"i"(n) : "memory")
#endif
#else
#define WAIT_ASYNC(n)
#endif

#if USE_ASYNC_LDS
// Async 16B copy global -> LDS (v4i addrspace(1) src, addrspace(3) dst).
__device__ __forceinline__ void async_cp_b128(const fvec4* g, fvec4* l) {
    __builtin_amdgcn_global_load_async_to_lds_b128(
        (AS1 iv4*)(g), (AS3 iv4*)(l), 0, 0);
}
#endif

__device__ __forceinline__ fvec4 f4_fma(float s, fvec4 a, fvec4 c) {
    fvec4 r;
    r.x = fmaf(s, a.x, c.x);
    r.y = fmaf(s, a.y, c.y);
    r.z = fmaf(s, a.z, c.z);
    r.w = fmaf(s, a.w, c.w);
    return r;
}

// E_CT > 0: compile-time expert count (fully unrolled pipeline).
// E_CT == 0: runtime expert count fallback.
template <int E_CT>
__global__ __launch_bounds__(kBlock) void moe_reduce_rmsnorm_kernel(
    const float* __restrict__ residual,     // [T, H]
    const float* __restrict__ norm_weight,  // [H]
    const float* __restrict__ scale,        // [E, T]
    const float* __restrict__ act,          // [E, T, H]
    const float* __restrict__ token,        // [T, H]
    float* __restrict__ out_hidden,         // [T, H]  (d_out first half)
    float* __restrict__ out_resid,          // [T, H]  (d_out second half)
    int T, int E_rt) {
    const int E   = (E_CT > 0) ? E_CT : E_rt;
    const int t   = blockIdx.x;
    const int tid = threadIdx.x;
    const size_t rowOff = (size_t)t * kNV;

    const fvec4* res4 = (const fvec4*)residual + rowOff;
    const fvec4* tok4 = (const fvec4*)token + rowOff;
    const fvec4* wt4  = (const fvec4*)norm_weight;
    const fvec4* act4 = (const fvec4*)act;
    fvec4* hid4 = (fvec4*)out_hidden + rowOff;
    fvec4* orr4 = (fvec4*)out_resid + rowOff;

#if USE_ASYNC_LDS
    __shared__ fvec4 lbuf[2][kNV];  // 2 x 16 KB double buffer
    // Kick off the dominant stream (expert 0) FIRST so the async engine is
    // pulling from HBM while we load residual+token into registers.
    {
        const fvec4* g0 = act4 + (size_t)t * kNV;  // e = 0
#pragma unroll
        for (int c = 0; c < kChunks; ++c) {
            const int i = tid + c * kBlock;
            async_cp_b128(g0 + i, &lbuf[0][i]);
        }
    }
#endif

    // ---- residual + token (streamed once -> non-temporal b128 loads) ----
    fvec4 acc[kChunks];
#pragma unroll
    for (int c = 0; c < kChunks; ++c) {
        const int i = tid + c * kBlock;
        fvec4 r = __builtin_nontemporal_load(&res4[i]);
        fvec4 v = __builtin_nontemporal_load(&tok4[i]);
        acc[c] = r + v;
    }

    // ---- expert reduction: acc += scale[e,t] * act[e,t,:] ----
#if USE_ASYNC_LDS
    // Each thread's async destinations are exactly the slots it reads back,
    // so no cross-wave barrier is needed; ASYNCcnt in-order completion
    // sequences the double buffer.
#pragma unroll
    for (int e = 0; e < E; ++e) {
        const float s = scale[(size_t)e * T + t];  // block-uniform -> s_load
        if (e + 1 < E) {
            const fvec4* g = act4 + ((size_t)(e + 1) * T + t) * kNV;
            const int b = (e + 1) & 1;
#pragma unroll
            for (int c = 0; c < kChunks; ++c) {
                const int i = tid + c * kBlock;
                async_cp_b128(g + i, &lbuf[b][i]);
            }
            WAIT_ASYNC(kChunks);  // oldest kChunks (current buffer) complete
        } else {
            WAIT_ASYNC(0);
        }
        asm volatile("" ::: "memory");
#pragma unroll
        for (int c = 0; c < kChunks; ++c) {
            const int i = tid + c * kBlock;
            acc[c] = f4_fma(s, lbuf[e & 1][i], acc[c]);
        }
    }
#else
#pragma unroll
    for (int e = 0; e < E; ++e) {
        const float s = scale[(size_t)e * T + t];
        const fvec4* g = act4 + ((size_t)e * T + t) * kNV;
#pragma unroll
        for (int c = 0; c < kChunks; ++c) {
            const int i = tid + c * kBlock;
            fvec4 a = __builtin_nontemporal_load(&g[i]);
            acc[c] = f4_fma(s, a, acc[c]);
        }
    }
#endif

    // ---- write output_residual (never re-read -> non-temporal store),
    //      accumulate per-thread sum of squares ----
    float ss = 0.f;
#pragma unroll
    for (int c = 0; c < kChunks; ++c) {
        const int i = tid + c * kBlock;
        __builtin_nontemporal_store(acc[c], &orr4[i]);
        ss = fmaf(acc[c].x, acc[c].x, ss);
        ss = fmaf(acc[c].y, acc[c].y, ss);
        ss = fmaf(acc[c].z, acc[c].z, ss);
        ss = fmaf(acc[c].w, acc[c].w, ss);
    }

    // ---- block reduction: wave32 shuffle tree + LDS across 8 waves ----
#pragma unroll
    for (int off = 16; off > 0; off >>= 1) ss += __shfl_xor(ss, off, 32);
    __shared__ float red[kWaves];
    if ((tid & 31) == 0) red[tid >> 5] = ss;
    __syncthreads();
    float tot = 0.f;
#pragma unroll
    for (int w = 0; w < kWaves; ++w) tot += red[w];

    const float inv = rsqrtf(tot * (1.0f / (float)kH) + kEps);

    // ---- normalize + weight; norm_weight via cached (RT) loads ----
#pragma unroll
    for (int c = 0; c < kChunks; ++c) {
        const int i = tid + c * kBlock;
        fvec4 w = wt4[i];
        fvec4 h;
        h.x = acc[c].x * inv * w.x;
        h.y = acc[c].y * inv * w.y;
        h.z = acc[c].z * inv * w.z;
        h.w = acc[c].w * inv * w.w;
        __builtin_nontemporal_store(h, &hid4[i]);
    }
}

extern "C" void kernel_launch(void* const* d_in, const int* in_sizes, int n_in,
                              void* d_out, int out_size, void* d_ws, size_t ws_size,
                              hipStream_t stream) {
    (void)n_in; (void)out_size; (void)d_ws; (void)ws_size;
    const int H = in_sizes[1];       // 4096
    const int T = in_sizes[0] / H;   // 8192
    const int E = in_sizes[2] / T;   // 8

    const float* residual    = (const float*)d_in[0];
    const float* norm_weight = (const float*)d_in[1];
    const float* scale       = (const float*)d_in[2];
    const float* act         = (const float*)d_in[3];
    const float* token       = (const float*)d_in[4];

    float* out_hidden = (float*)d_out;                      // hidden_states
    float* out_resid  = (float*)d_out + (size_t)T * H;      // output_residual

    if (E == 8) {
        moe_reduce_rmsnorm_kernel<8><<<dim3(T), dim3(kBlock), 0, stream>>>(
            residual, norm_weight, scale, act, token, out_hidden, out_resid, T, E);
    } else {
        moe_reduce_rmsnorm_kernel<0><<<dim3(T), dim3(kBlock), 0, stream>>>(
            residual, norm_weight, scale, act, token, out_hidden, out_resid, T, E);
    }
}